// EdgeGCNLayer_30442728194386
// MI455X (gfx1250) — compile-verified
//
#include <hip/hip_runtime.h>
#include <hip/hip_bf16.h>

#define N_NODES 20000
#define N_EDGES 640000
#define FEATS   128

typedef __attribute__((ext_vector_type(2))) float v2f;
typedef __attribute__((ext_vector_type(8))) float v8f;

// ---------------------------------------------------------------------------
// Kernel 1: zero the accumulator (h_neigh) and degree arrays in workspace.
// ---------------------------------------------------------------------------
__global__ __launch_bounds__(256)
void egcn_zero_kernel(float* __restrict__ p, int n) {
    int i = blockIdx.x * 256 + threadIdx.x;
    if (i < n) p[i] = 0.0f;
}

// ---------------------------------------------------------------------------
// Kernel 2: segment-sum of edge features into acc[dst], and degree counts.
// 32 lanes cooperate on one edge; each lane handles 4 consecutive feats via a
// float4 load (512B coalesced per wave) and 4 hardware fp32 atomics. The
// 10 MB accumulator lives in L2 (192 MB), so atomics never touch HBM.
// ---------------------------------------------------------------------------
__global__ __launch_bounds__(256)
void egcn_scatter_kernel(const float* __restrict__ ef,
                         const int*   __restrict__ dst,
                         float* __restrict__ acc,
                         float* __restrict__ deg) {
    const int gid = blockIdx.x * 256 + threadIdx.x;
    const int e   = gid >> 5;            // edge index (32 lanes per edge)
    const int q   = (gid & 31) << 2;     // feature offset: 0,4,...,124
    const int d   = dst[e];
    const float4 v = *(const float4*)(ef + (size_t)e * FEATS + q);
    float* row = acc + (size_t)d * FEATS + q;
    unsafeAtomicAdd(row + 0, v.x);
    unsafeAtomicAdd(row + 1, v.y);
    unsafeAtomicAdd(row + 2, v.z);
    unsafeAtomicAdd(row + 3, v.w);
    if ((gid & 31) == 0) unsafeAtomicAdd(deg + d, 1.0f);
}

// ---------------------------------------------------------------------------
// Kernel 3: OUT = (ACC @ W^T + b) * rsqrt(max(deg,1)) using fp32 WMMA.
// One wave32 per 16x16 output tile; K=128 -> 32 x V_WMMA_F32_16X16X4_F32.
//
// A (16x4, MxK) VGPR layout: lanes 0-15 = rows M=0..15 with (K=0,K=1) in the
// two VGPRs; lanes 16-31 = same rows with (K=2,K=3).  B (4x16, KxN) mirrors
// this with lanes carrying N.  C/D: VGPR r holds M=r (lanes 0-15, N=lane) and
// M=r+8 (lanes 16-31, N=lane-16).
// ---------------------------------------------------------------------------
__global__ __launch_bounds__(32)
void egcn_gemm_norm_kernel(const float* __restrict__ acc,
                           const float* __restrict__ W,     // [128 out][128 in]
                           const float* __restrict__ bias,  // [128]
                           const float* __restrict__ deg,   // [N_NODES]
                           float* __restrict__ out) {       // [N_NODES][128]
    const int lane = threadIdx.x & 31;
    const int half = lane >> 4;          // 0: lanes 0-15, 1: lanes 16-31
    const int lm   = lane & 15;
    const int tile = blockIdx.x;
    const int n0   = (tile >> 3) << 4;   // node-row tile base (1250 tiles)
    const int o0   = (tile & 7)  << 4;   // out-col tile base  (8 tiles)

    // Per-lane base pointers: +2*half selects the K pair this lane carries.
    const float* aptr = acc + (size_t)(n0 + lm) * FEATS + 2 * half;
    // OUT = ACC @ W^T  =>  B[k][o] = W[o][k]; lane carries N=o0+lm.
    const float* bptr = W   + (size_t)(o0 + lm) * FEATS + 2 * half;

    v8f c = {};
#pragma unroll
    for (int kk = 0; kk < FEATS; kk += 4) {
        v2f a = *(const v2f*)(aptr + kk);
        v2f b = *(const v2f*)(bptr + kk);
        c = __builtin_amdgcn_wmma_f32_16x16x4_f32(
                /*neg_a=*/false, a, /*neg_b=*/false, b,
                /*c_mod=*/(short)0, c, /*reuse_a=*/false, /*reuse_b=*/false);
    }

    const float bv = bias[o0 + lm];
#pragma unroll
    for (int r = 0; r < 8; ++r) {
        const int row = n0 + r + 8 * half;
        float dv = deg[row];
        dv = dv < 1.0f ? 1.0f : dv;
        const float s = rsqrtf(dv);
        out[(size_t)row * FEATS + (o0 + lm)] = (c[r] + bv) * s;
    }
}

// ---------------------------------------------------------------------------
// Host launcher
// ---------------------------------------------------------------------------
extern "C" void kernel_launch(void* const* d_in, const int* in_sizes, int n_in,
                              void* d_out, int out_size, void* d_ws, size_t ws_size,
                              hipStream_t stream) {
    // setup_inputs order: node_feats, edge_feats, W, b, src, dst
    const float* edge_feats = (const float*)d_in[1];
    const float* W          = (const float*)d_in[2];
    const float* bias       = (const float*)d_in[3];
    const int*   dst        = (const int*)  d_in[5];
    float* out = (float*)d_out;

    // Workspace: acc [N_NODES*FEATS] floats, then deg [N_NODES] floats.
    float* acc = (float*)d_ws;
    float* deg = acc + (size_t)N_NODES * FEATS;

    const int ztot = N_NODES * FEATS + N_NODES;                 // 2,580,000
    egcn_zero_kernel<<<(ztot + 255) / 256, 256, 0, stream>>>(acc, ztot);

    const int scatter_blocks = (N_EDGES * 32) / 256;            // 80,000
    egcn_scatter_kernel<<<scatter_blocks, 256, 0, stream>>>(edge_feats, dst, acc, deg);

    const int gemm_blocks = (N_NODES / 16) * (FEATS / 16);      // 10,000
    egcn_gemm_norm_kernel<<<gemm_blocks, 32, 0, stream>>>(acc, W, bias, deg, out);
}